// IDRate_86852828660212
// MI455X (gfx1250) — compile-verified
//
#include <hip/hip_runtime.h>
#include <stdint.h>

#define C_CLS    26
#define NVOX     (96 * 96 * 96)       // 884736 voxels per (batch, channel)
#define CHUNK    1024                 // voxels per block
#define NCHUNK   (NVOX / CHUNK)       // 864
#define BLOCK    256                  // 8 wave32; 4 voxels (16B) per thread
#define CH_BYTES 4096                 // CHUNK * 4 bytes, one channel slice in LDS
#define DEPTH    8                    // async pipeline depth (channels in flight)

static_assert(C_CLS - DEPTH == 18, "drain sequence below assumes 26 channels, depth 8");

template <int N>
__device__ __forceinline__ void wait_async() {
    asm volatile("s_wait_asynccnt %0" :: "n"(N) : "memory");
}

// ---------------------------------------------------------------- init ws
__global__ void idrate_init(int* ws) {
    int i = threadIdx.x;
    if (i < 2 * C_CLS * 2) ws[i] = 0;
}

// ---------------------------------------------------------------- main
__global__ __launch_bounds__(BLOCK) void idrate_main(
    const float* __restrict__ pred, const float* __restrict__ targ,
    int* __restrict__ g_total, int* __restrict__ g_correct)
{
    // dynamic LDS: [DEPTH stages][tensor 0=pred,1=targ][4KB] + histograms
    extern __shared__ char smem[];
    int* s_total   = (int*)(smem + DEPTH * 2 * CH_BYTES);
    int* s_correct = s_total + C_CLS;

    const int tid = threadIdx.x;
    const int b   = blockIdx.y;

    if (tid < C_CLS) { s_total[tid] = 0; s_correct[tid] = 0; }

    // this lane's 16B segment base in LDS (low 32 bits of generic LDS ptr)
    const uint32_t lds_lane = (uint32_t)(uintptr_t)smem + (uint32_t)tid * 16u;
    // uniform 64-bit batch bases for GVS-mode async loads
    const uint64_t pbase = (uint64_t)(uintptr_t)(pred + (size_t)b * C_CLS * NVOX);
    const uint64_t tbase = (uint64_t)(uintptr_t)(targ + (size_t)b * C_CLS * NVOX);
    // per-lane 32-bit byte offset within the batch (max ~92MB, fits signed 32-bit)
    const uint32_t goff0 = (uint32_t)blockIdx.x * (CHUNK * 4u) + (uint32_t)tid * 16u;

    // issue one stage (channel c): two b128 async DMAs into slot c % DEPTH
    auto issue = [&](int c) {
        uint32_t ldsP = lds_lane + ((uint32_t)c & (DEPTH - 1u)) * (2u * CH_BYTES);
        uint32_t ldsT = ldsP + CH_BYTES;
        uint32_t goff = goff0 + (uint32_t)c * (NVOX * 4u);
        asm volatile("global_load_async_to_lds_b128 %0, %1, %2"
                     :: "v"(ldsP), "v"(goff), "s"(pbase) : "memory");
        asm volatile("global_load_async_to_lds_b128 %0, %1, %2"
                     :: "v"(ldsT), "v"(goff), "s"(tbase) : "memory");
    };

    float bestP[4], bestT[4];
    int   idxP[4],  idxT[4];
#pragma unroll
    for (int k = 0; k < 4; ++k) {
        bestP[k] = -__builtin_inff(); bestT[k] = -__builtin_inff();
        idxP[k] = 0; idxT[k] = 0;
    }

    // consume stage c from this lane's private LDS slot (no cross-wave sharing
    // -> per-wave s_wait_asynccnt is sufficient synchronization, no barriers)
    auto consume = [&](int c) {
        const char* bp = smem + ((c & (DEPTH - 1)) * (2 * CH_BYTES));
        const float4 vp = *(const float4*)(bp + tid * 16);
        const float4 vt = *(const float4*)(bp + CH_BYTES + tid * 16);
        // lane's ds reads retired -> slot reusable by the async engine
        asm volatile("s_wait_dscnt 0x0" ::: "memory");
        const float pv[4] = { vp.x, vp.y, vp.z, vp.w };
        const float tv[4] = { vt.x, vt.y, vt.z, vt.w };
#pragma unroll
        for (int k = 0; k < 4; ++k) {
            if (pv[k] > bestP[k]) { bestP[k] = pv[k]; idxP[k] = c; }  // first-max wins
            if (tv[k] > bestT[k]) { bestT[k] = tv[k]; idxT[k] = c; }
        }
    };

    // ---- prologue: fill the pipeline (16 async loads in flight per wave)
#pragma unroll
    for (int c = 0; c < DEPTH; ++c) issue(c);

    // ---- steady state: 16 in flight; wait<=14 completes stage c (in-order)
#pragma unroll
    for (int c = 0; c < C_CLS - DEPTH; ++c) {   // c = 0..17
        wait_async<2 * (DEPTH - 1)>();          // <= 14
        consume(c);
        issue(c + DEPTH);
    }

    // ---- drain: stages 18..25, in-flight shrinks by 2 each step
    wait_async<14>(); consume(18);
    wait_async<12>(); consume(19);
    wait_async<10>(); consume(20);
    wait_async<8>();  consume(21);
    wait_async<6>();  consume(22);
    wait_async<4>();  consume(23);
    wait_async<2>();  consume(24);
    wait_async<0>();  consume(25);

    // ---- block-local histograms (integer-exact), one barrier pair total
    __syncthreads();   // s_total/s_correct init visible to all waves
#pragma unroll
    for (int k = 0; k < 4; ++k) {
        atomicAdd(&s_total[idxT[k]], 1);
        if (idxP[k] == idxT[k]) atomicAdd(&s_correct[idxT[k]], 1);
    }
    __syncthreads();
    if (tid < C_CLS) {
        atomicAdd(&g_total[b * C_CLS + tid],   s_total[tid]);
        atomicAdd(&g_correct[b * C_CLS + tid], s_correct[tid]);
    }
}

// ---------------------------------------------------------------- finalize
__global__ void idrate_final(const int* __restrict__ g_total,
                             const int* __restrict__ g_correct,
                             float* __restrict__ out)
{
    if (threadIdx.x == 0 && blockIdx.x == 0) {
        float acc = 0.0f; int nvalid = 0;
        for (int b = 0; b < 2; ++b) {
            int np = 0; float sr = 0.0f;
            for (int c = 1; c < C_CLS; ++c) {        // class 0 excluded
                int t = g_total[b * C_CLS + c];
                if (t > 0) {
                    ++np;
                    sr += (float)g_correct[b * C_CLS + c] / (float)t;
                }
            }
            if (np > 0) { ++nvalid; acc += sr / (float)np; }
        }
        out[0] = (nvalid > 0) ? (acc / (float)nvalid) : 0.0f;
    }
}

// ---------------------------------------------------------------- launch
extern "C" void kernel_launch(void* const* d_in, const int* in_sizes, int n_in,
                              void* d_out, int out_size, void* d_ws, size_t ws_size,
                              hipStream_t stream) {
    const float* pred = (const float*)d_in[0];
    const float* targ = (const float*)d_in[1];
    int* g_total   = (int*)d_ws;                 // [2][26]
    int* g_correct = g_total + 2 * C_CLS;        // [2][26]

    idrate_init<<<1, 128, 0, stream>>>(g_total);

    dim3 grid(NCHUNK, 2);
    size_t shmem = DEPTH * 2 * CH_BYTES + 2 * C_CLS * sizeof(int);
    idrate_main<<<grid, BLOCK, shmem, stream>>>(pred, targ, g_total, g_correct);

    idrate_final<<<1, 1, 0, stream>>>(g_total, g_correct, (float*)d_out);
}